// SequenceAttention_28733331210483
// MI455X (gfx1250) — compile-verified
//
#include <hip/hip_runtime.h>
#include <hip/hip_bf16.h>
#include <math.h>

typedef __attribute__((ext_vector_type(2))) float v2f;
typedef __attribute__((ext_vector_type(8))) float v8f;

constexpr int B_ = 2, L_ = 512, CS = 384, CZ = 128, H_ = 8, HD = 16;
constexpr float EPS_ = 1e-5f;
constexpr float THETA_ = 10000.0f;
constexpr float NEGINF = -3.402823466e38f;

// ---------------- Kernel 1: x = rmsnorm(s@Ws^T)*gs ; qk = x@Wqk^T ; RoPE -> q,k ----
__global__ void __launch_bounds__(128)
prep_qk_kernel(const float* __restrict__ s, const float* __restrict__ Ws,
               const float* __restrict__ gs, const float* __restrict__ Wqk,
               float* __restrict__ qout, float* __restrict__ kout) {
  __shared__ float sh_s[CS];
  __shared__ float sh_x[CZ];
  __shared__ float sh_q[CZ];
  __shared__ float sh_k[CZ];
  __shared__ float sh_red[4];

  const int bl  = blockIdx.x;        // b*L + l
  const int l   = bl & (L_ - 1);
  const int tid = threadIdx.x;       // 0..127 (one channel each)

  // load s row as float4 (96 of them)
  {
    const float4* srow = (const float4*)(s + (size_t)bl * CS);
    if (tid < CS / 4) ((float4*)sh_s)[tid] = srow[tid];
  }
  __syncthreads();

  // y[c] = dot(s_row, Ws[c,:])   (b128 weight loads)
  float y = 0.f;
  {
    const float4* wr = (const float4*)(Ws + (size_t)tid * CS);
    #pragma unroll 4
    for (int j = 0; j < CS / 4; ++j) {
      float4 w = wr[j];
      y = fmaf(sh_s[4 * j + 0], w.x, y);
      y = fmaf(sh_s[4 * j + 1], w.y, y);
      y = fmaf(sh_s[4 * j + 2], w.z, y);
      y = fmaf(sh_s[4 * j + 3], w.w, y);
    }
  }
  // rmsnorm over 128 channels
  float ss = y * y;
  #pragma unroll
  for (int off = 16; off; off >>= 1) ss += __shfl_xor(ss, off, 32);
  if ((tid & 31) == 0) sh_red[tid >> 5] = ss;
  __syncthreads();
  float tot = sh_red[0] + sh_red[1] + sh_red[2] + sh_red[3];
  float x = y * rsqrtf(tot * (1.0f / CZ) + EPS_) * gs[tid];
  sh_x[tid] = x;
  __syncthreads();

  // qk = x @ Wqk^T  (rows 0..127 -> q, 128..255 -> k)
  float qv = 0.f, kv = 0.f;
  {
    const float4* wq = (const float4*)(Wqk + (size_t)tid * CZ);
    const float4* wk = (const float4*)(Wqk + (size_t)(tid + CZ) * CZ);
    #pragma unroll 4
    for (int j = 0; j < CZ / 4; ++j) {
      float4 a = wq[j], b = wk[j];
      qv = fmaf(sh_x[4 * j + 0], a.x, qv);  kv = fmaf(sh_x[4 * j + 0], b.x, kv);
      qv = fmaf(sh_x[4 * j + 1], a.y, qv);  kv = fmaf(sh_x[4 * j + 1], b.y, kv);
      qv = fmaf(sh_x[4 * j + 2], a.z, qv);  kv = fmaf(sh_x[4 * j + 2], b.z, kv);
      qv = fmaf(sh_x[4 * j + 3], a.w, qv);  kv = fmaf(sh_x[4 * j + 3], b.w, kv);
    }
  }
  sh_q[tid] = qv;
  sh_k[tid] = kv;
  __syncthreads();

  // RoPE (interleaved pairs, rotary_dim = hd = 16)
  const int d  = tid & (HD - 1);
  const int jj = d >> 1;
  const float inv = powf(THETA_, -(float)jj * (1.0f / 8.0f));
  const float ang = (float)l * inv;
  const float cs = cosf(ang), sn = sinf(ang);
  float qo, ko;
  if ((d & 1) == 0) {
    qo = sh_q[tid] * cs - sh_q[tid + 1] * sn;
    ko = sh_k[tid] * cs - sh_k[tid + 1] * sn;
  } else {
    qo = sh_q[tid - 1] * sn + sh_q[tid] * cs;
    ko = sh_k[tid - 1] * sn + sh_k[tid] * cs;
  }
  qout[(size_t)bl * CZ + tid] = qo;
  kout[(size_t)bl * CZ + tid] = ko;
}

// ---------------- Kernel 2: scores + softmax + (P @ Wo^T) via WMMA + rmsnorm ------
__global__ void __launch_bounds__(256)
attn_out_kernel(const float* __restrict__ qin, const float* __restrict__ kin,
                const int* __restrict__ mask, const float* __restrict__ Wo,
                const float* __restrict__ gz, float* __restrict__ out) {
  __shared__ float sh_q[CZ];
  __shared__ float sh_p[H_][L_];     // attention probs p[h][m]

  const int bl   = blockIdx.x;       // b*L + l
  const int b    = bl >> 9;
  const int tid  = threadIdx.x;      // 0..255  (8 wave32s)
  const int lane = tid & 31;
  const int wave = tid >> 5;

  if (tid < CZ) sh_q[tid] = qin[(size_t)bl * CZ + tid];
  __syncthreads();

  const int mask_l = mask[bl];

  // --- scores: p[h][m] = (q[h,:]·k[m,h,:]) / sqrt(hd), masked ---
  for (int m = tid; m < L_; m += 256) {
    const float4* krow = (const float4*)(kin + (size_t)(b * L_ + m) * CZ);
    const int mm = mask_l & mask[b * L_ + m];
    #pragma unroll
    for (int h = 0; h < H_; ++h) {
      float dot = 0.f;
      #pragma unroll
      for (int v = 0; v < 4; ++v) {              // 4 x float4 = 16 elems
        float4 kv = krow[h * 4 + v];
        dot = fmaf(sh_q[h * HD + 4 * v + 0], kv.x, dot);
        dot = fmaf(sh_q[h * HD + 4 * v + 1], kv.y, dot);
        dot = fmaf(sh_q[h * HD + 4 * v + 2], kv.z, dot);
        dot = fmaf(sh_q[h * HD + 4 * v + 3], kv.w, dot);
      }
      sh_p[h][m] = mm ? dot * 0.25f : NEGINF;
    }
  }
  __syncthreads();

  // --- softmax along m: wave w owns head h = w ---
  {
    const int h = wave;
    float mx = NEGINF;
    for (int m = lane; m < L_; m += 32) mx = fmaxf(mx, sh_p[h][m]);
    #pragma unroll
    for (int off = 16; off; off >>= 1) mx = fmaxf(mx, __shfl_xor(mx, off, 32));
    float sum = 0.f;
    for (int m = lane; m < L_; m += 32) {
      float e = __expf(sh_p[h][m] - mx);
      sh_p[h][m] = e;
      sum += e;
    }
    #pragma unroll
    for (int off = 16; off; off >>= 1) sum += __shfl_xor(sum, off, 32);
    const float isum = 1.0f / sum;
    for (int m = lane; m < L_; m += 32) sh_p[h][m] *= isum;
  }
  __syncthreads();

  // --- output: out[m][c] = sum_h p[h][m] * Wo[c][h]; then rmsnorm row, write ---
  // WMMA f32 16x16x4, two chained calls cover K=H=8.
  // Tile-column -> channel remap: tile ct, column n  ->  channel n*8 + ct.
  // Thus lane ln's 8 accumulators are 8 CONSECUTIVE channels ln*8..ln*8+7,
  // enabling two global_store_b128 per row per lane.
  const int half = lane >> 4;        // 0: lanes 0-15, 1: lanes 16-31
  const int ln   = lane & 15;

  // B fragments (4x16 f32): VGPR0 = rows {0|2}, VGPR1 = rows {1|3} per half.
  v2f bA[8], bB[8];
  float gzv[8];
  #pragma unroll
  for (int ct = 0; ct < 8; ++ct) {
    const int c  = ln * 8 + ct;           // physical output channel for column ln
    const int k0 = half ? 2 : 0;          // heads 0..3 pass
    const int k1 = half ? 6 : 4;          // heads 4..7 pass
    bA[ct] = (v2f){ Wo[c * H_ + k0], Wo[c * H_ + k0 + 1] };
    bB[ct] = (v2f){ Wo[c * H_ + k1], Wo[c * H_ + k1 + 1] };
    gzv[ct] = gz[c];
  }

  const size_t outbase = (size_t)bl * L_ * CZ;

  #pragma unroll
  for (int t = 0; t < 4; ++t) {
    const int mt = wave + t * 8;          // m-tile index (32 tiles of 16 rows)
    const int m  = mt * 16 + ln;          // A-matrix row for this lane

    // A fragments (16x4 f32): lanes 0-15 -> K={0,1}, lanes 16-31 -> K={2,3}
    const int h0 = half ? 2 : 0;
    const int h1 = half ? 6 : 4;
    v2f aA = (v2f){ sh_p[h0][m], sh_p[h0 + 1][m] };
    v2f aB = (v2f){ sh_p[h1][m], sh_p[h1 + 1][m] };

    v8f acc[8];
    #pragma unroll
    for (int ct = 0; ct < 8; ++ct) {
      v8f zero = {0.f, 0.f, 0.f, 0.f, 0.f, 0.f, 0.f, 0.f};
      acc[ct] = __builtin_amdgcn_wmma_f32_16x16x4_f32(
          false, aA, false, bA[ct], (short)0, zero, false, false);
      acc[ct] = __builtin_amdgcn_wmma_f32_16x16x4_f32(
          false, aB, false, bB[ct], (short)0, acc[ct], false, false);
    }

    // rmsnorm each of the 16 output rows; C/D layout: VGPR r -> row r (+8 upper half)
    #pragma unroll
    for (int r = 0; r < 8; ++r) {
      float ssq = 0.f;
      #pragma unroll
      for (int ct = 0; ct < 8; ++ct) ssq = fmaf(acc[ct][r], acc[ct][r], ssq);
      // reduce across the 16-lane half (masks 1,2,4,8 stay within the half)
      #pragma unroll
      for (int off = 8; off; off >>= 1) ssq += __shfl_xor(ssq, off, 32);
      const float scale = rsqrtf(ssq * (1.0f / CZ) + EPS_);
      const int mrow = mt * 16 + r + half * 8;

      float4 lo, hi;
      lo.x = acc[0][r] * scale * gzv[0];
      lo.y = acc[1][r] * scale * gzv[1];
      lo.z = acc[2][r] * scale * gzv[2];
      lo.w = acc[3][r] * scale * gzv[3];
      hi.x = acc[4][r] * scale * gzv[4];
      hi.y = acc[5][r] * scale * gzv[5];
      hi.z = acc[6][r] * scale * gzv[6];
      hi.w = acc[7][r] * scale * gzv[7];

      float4* orow = (float4*)(out + outbase + (size_t)mrow * CZ + ln * 8);
      orow[0] = lo;
      orow[1] = hi;
    }
  }
}

extern "C" void kernel_launch(void* const* d_in, const int* in_sizes, int n_in,
                              void* d_out, int out_size, void* d_ws, size_t ws_size,
                              hipStream_t stream) {
  const float* s    = (const float*)d_in[0];
  const int*   mask = (const int*)d_in[1];
  const float* Ws   = (const float*)d_in[2];
  const float* gs   = (const float*)d_in[3];
  const float* Wqk  = (const float*)d_in[4];
  const float* Wo   = (const float*)d_in[5];
  const float* gz   = (const float*)d_in[6];
  float* out = (float*)d_out;

  float* q = (float*)d_ws;                        // B*L*CZ f32 = 512 KB
  float* k = q + (size_t)B_ * L_ * CZ;            // next 512 KB

  prep_qk_kernel<<<B_ * L_, 128, 0, stream>>>(s, Ws, gs, Wqk, q, k);
  attn_out_kernel<<<B_ * L_, 256, 0, stream>>>(q, k, mask, Wo, gz, out);
}